// CombinedModel_16097537425808
// MI455X (gfx1250) — compile-verified
//
#include <hip/hip_runtime.h>

typedef __attribute__((ext_vector_type(2))) float v2f;
typedef __attribute__((ext_vector_type(8))) float v8f;

#define HID 256

// ---------------------------------------------------------------------------
// Kernel A: sequential ODE-RNN recurrence. One wave (32 lanes).
// Lanes cooperatively load 32 timesteps of u (coalesced), broadcast each
// timestep via wave32 shuffles, all lanes redundantly advance the 3-dim
// state, lane i captures the state after step (t0+i), then the chunk of 32
// states per channel stores back coalesced.
// ---------------------------------------------------------------------------
__global__ __launch_bounds__(32)
void rnode_recurrence(const float* __restrict__ u, const float* __restrict__ dtp,
                      const float* __restrict__ A, const float* __restrict__ Bm,
                      const float* __restrict__ bA, float* __restrict__ mem, int S)
{
  const int lane = threadIdx.x;
  const float dt = dtp[0];
  const float a00 = A[0], a01 = A[1], a02 = A[2];
  const float a10 = A[3], a11 = A[4], a12 = A[5];
  const float a20 = A[6], a21 = A[7], a22 = A[8];
  const float c00 = Bm[0], c01 = Bm[1], c02 = Bm[2];
  const float c10 = Bm[3], c11 = Bm[4], c12 = Bm[5];
  const float c20 = Bm[6], c21 = Bm[7], c22 = Bm[8];
  const float p0 = bA[0], p1 = bA[1], p2 = bA[2];

  float x0 = 0.f, x1 = 0.f, x2 = 0.f;

  for (int t0 = 0; t0 < S; t0 += 32) {
    // coalesced chunk load of u[:, t0..t0+31]
    float u0 = u[0 * S + t0 + lane];
    float u1 = u[1 * S + t0 + lane];
    float u2 = u[2 * S + t0 + lane];
    float m0 = 0.f, m1 = 0.f, m2 = 0.f;
    #pragma unroll 1
    for (int i = 0; i < 32; ++i) {
      const float s0v = __shfl(u0, i, 32);
      const float s1v = __shfl(u1, i, 32);
      const float s2v = __shfl(u2, i, 32);
      const float z0 = fmaf(a00, x0, fmaf(a01, x1, fmaf(a02, x2,
                       fmaf(c00, s0v, fmaf(c01, s1v, fmaf(c02, s2v, p0))))));
      const float z1 = fmaf(a10, x0, fmaf(a11, x1, fmaf(a12, x2,
                       fmaf(c10, s0v, fmaf(c11, s1v, fmaf(c12, s2v, p1))))));
      const float z2 = fmaf(a20, x0, fmaf(a21, x1, fmaf(a22, x2,
                       fmaf(c20, s0v, fmaf(c21, s1v, fmaf(c22, s2v, p2))))));
      x0 = fmaf(dt, tanhf(z0), x0);
      x1 = fmaf(dt, tanhf(z1), x1);
      x2 = fmaf(dt, tanhf(z2), x2);
      if (lane == i) { m0 = x0; m1 = x1; m2 = x2; }
    }
    // coalesced chunk store of membrane potentials
    mem[0 * S + t0 + lane] = m0;
    mem[1 * S + t0 + lane] = m1;
    mem[2 * S + t0 + lane] = m2;
  }
}

// ---------------------------------------------------------------------------
// Kernel B: per-timestep MLP via fp32 WMMA. One wave handles 16 timesteps.
//   GEMM1: h[256,16] = relu(W1[256,3] @ x[3,16] + b1)  -> 16x wmma (K=3 pad 4)
//   GEMM2: out[3,16] = W2[3,256] @ h + b2              -> 64x wmma (K=4 steps)
// f32 WMMA operand layout (per ISA): A/B element k lives at vgpr (k%2),
// lane-half (k/2); lane%16 = M (A) or N (B). C/D: vgpr j <-> row j+8*half.
// All pad-lane loads are done unconditionally in-bounds (clamped index) and
// zeroed with a value-select, so no exec-mask divergence interleaves with
// the WMMA stream.
// ---------------------------------------------------------------------------
__global__ __launch_bounds__(32)
void mlp_wmma(const float* __restrict__ mem, const float* __restrict__ W1,
              const float* __restrict__ b1, const float* __restrict__ W2,
              const float* __restrict__ b2, float* __restrict__ out, int S)
{
  __shared__ float hs[16][HID + 4];   // h tile, column-major, padded stride

  const int lane = threadIdx.x;
  const int half = lane >> 4;         // 0: lanes 0-15, 1: lanes 16-31
  const int col  = lane & 15;         // column within the 16-wide tile
  const int s0   = blockIdx.x * 16;   // first timestep of this tile

  // B operand: x tile [4x16], row 3 zero-padded. k = vgpr + 2*half.
  // Both loads are in-bounds for every lane; pad is a value-select.
  const float bx_hi = mem[1 * S + s0 + col];
  v2f bx;
  bx.x = mem[(2 * half) * S + s0 + col];                 // k = 0 or 2
  bx.y = half ? 0.0f : bx_hi;                            // k = 1 or 3(pad)

  // GEMM1: 16 independent 16x16 h-tiles, bias preloaded into C.
  v8f acc[16];
  #pragma unroll
  for (int m = 0; m < 16; ++m) {
    const int row = 16 * m + col;                        // W1 row (M index)
    const float w_mid = W1[row * 3 + 1];                 // always in-bounds
    v2f a;
    a.x = W1[row * 3 + 2 * half];                        // k = 0 or 2
    a.y = half ? 0.0f : w_mid;                           // k = 1 or 3(pad)
    v8f c;
    #pragma unroll
    for (int j = 0; j < 8; ++j) c[j] = b1[16 * m + 8 * half + j];
    acc[m] = __builtin_amdgcn_wmma_f32_16x16x4_f32(
        false, a, false, bx, (short)0, c, false, false);
  }

  // ReLU and stash to LDS column-major (8 contiguous floats per lane/tile).
  #pragma unroll
  for (int m = 0; m < 16; ++m) {
    float* dst = &hs[col][16 * m + 8 * half];
    #pragma unroll
    for (int j = 0; j < 8; ++j) dst[j] = fmaxf(acc[m][j], 0.0f);
  }
  __syncthreads();

  // GEMM2: accumulate 64 K=4 chunks; C preloaded with b2 (rows 0..2 valid).
  const int ccl = (col < 3) ? col : 2;                   // clamped, in-bounds
  const float amask = (col < 3) ? 1.0f : 0.0f;
  v8f c2;
  #pragma unroll
  for (int j = 0; j < 8; ++j) {
    const int r = j + 8 * half;
    const float bv = b2[(r < 3) ? r : 2];                // in-bounds load
    c2[j] = (r < 3) ? bv : 0.0f;
  }
  #pragma unroll 8
  for (int kk = 0; kk < 64; ++kk) {
    const int k0 = 4 * kk + 2 * half;
    v2f a2, bb;
    a2.x = amask * W2[ccl * HID + k0];                   // A row = out channel
    a2.y = amask * W2[ccl * HID + k0 + 1];
    bb.x = hs[col][k0];                                  // B from LDS
    bb.y = hs[col][k0 + 1];
    c2 = __builtin_amdgcn_wmma_f32_16x16x4_f32(
        false, a2, false, bb, (short)0, c2, false, false);
  }

  // D rows 0..2 (lanes 0-15, vgprs 0-2) are the outputs.
  if (half == 0) {
    #pragma unroll
    for (int j = 0; j < 3; ++j) out[j * S + s0 + col] = c2[j];
  }
}

// ---------------------------------------------------------------------------
extern "C" void kernel_launch(void* const* d_in, const int* in_sizes, int n_in,
                              void* d_out, int out_size, void* d_ws, size_t ws_size,
                              hipStream_t stream) {
  const float* u  = (const float*)d_in[0];
  const float* dt = (const float*)d_in[1];
  const float* A  = (const float*)d_in[2];
  const float* B  = (const float*)d_in[3];
  const float* bA = (const float*)d_in[4];
  const float* W1 = (const float*)d_in[5];
  const float* b1 = (const float*)d_in[6];
  const float* W2 = (const float*)d_in[7];
  const float* b2 = (const float*)d_in[8];

  float* out = (float*)d_out;
  const int S = in_sizes[0] / 3;          // 262144
  float* mem = out + (size_t)3 * S;       // membrane potentials region

  rnode_recurrence<<<1, 32, 0, stream>>>(u, dt, A, B, bA, mem, S);
  mlp_wmma<<<S / 16, 32, 0, stream>>>(mem, W1, b1, W2, b2, out, S);
}